// AdvancedGraphNeuralNetwork_78142634983587
// MI455X (gfx1250) — compile-verified
//
#include <hip/hip_runtime.h>
#include <stdint.h>

#define HID 128
#define NHEAD 8
#define DHEAD 16
#define NLAYER 4
#define NGRAPH 16
#define EPSV 1e-5f
#define NEG_SLOPE 0.2f
#define A_RS_PAD 4   // TDM pad: 4 dwords per row -> LDS row stride K+4 (bank-conflict free)

typedef __attribute__((ext_vector_type(2))) float    v2f;
typedef __attribute__((ext_vector_type(8))) float    v8f;
typedef __attribute__((ext_vector_type(4))) unsigned v4u;
typedef __attribute__((ext_vector_type(4))) int      v4i;
typedef __attribute__((ext_vector_type(8))) int      v8i;

__device__ __forceinline__ float lrelu(float x) { return x > 0.f ? x : NEG_SLOPE * x; }

// order-preserving float<->uint encoding so float max == uint max
__device__ __forceinline__ unsigned ord_enc(float f) {
    unsigned u = __float_as_uint(f);
    return (u & 0x80000000u) ? ~u : (u | 0x80000000u);
}
__device__ __forceinline__ float ord_dec(unsigned u) {
    return __uint_as_float((u & 0x80000000u) ? (u & 0x7fffffffu) : ~u);
}

// ---------------------------------------------------------------------------
// C[M,HID] = A[M,K] @ B[K,HID] (+bias). One wave per 16x16 tile, WMMA f32.
// The 16xK A-tile (shared by all 8 waves of the block) is DMA'd into LDS by
// the Tensor Data Mover (one tensor_load_to_lds per block, TENSORcnt-tracked),
// with D# row padding of 4 dwords so lane accesses hit distinct LDS banks.
// M must be a multiple of 16 (N=50000=16*3125); blockDim=256 = 8 waves = 8
// column tiles covering HID=128. No divergence at the WMMA (EXEC all ones).
// ---------------------------------------------------------------------------
__global__ void gemm_wmma_f32(const float* __restrict__ A, const float* __restrict__ B,
                              const float* __restrict__ bias, float* __restrict__ C, int K) {
    __shared__ float sA[16 * (HID + A_RS_PAD)];   // max K=128 -> 16*132 floats

    const int lane  = threadIdx.x & 31;
    const int wave  = threadIdx.x >> 5;           // column tile 0..7
    const int mbase = blockIdx.x * 16;
    const int col   = wave * 16 + (lane & 15);
    const int khalf = (lane >> 4) << 1;           // 0 (lanes 0-15) or 2 (lanes 16-31)
    const int RS    = K + A_RS_PAD;               // padded LDS row stride (floats)

    // --- TDM: DMA the 16xK f32 A-tile into LDS (wave 0 only, uniform branch) ---
    if (wave == 0) {
        const unsigned long long ga = (unsigned long long)(A + (size_t)mbase * K);
        const unsigned lds_base     = (unsigned)(uintptr_t)(&sA[0]);  // LDS offset = addr[31:0]
        // pad_interval code: interval = 2^(code+1) dwords; K=64 -> 5, K=128 -> 6
        const unsigned padcode = (K == 64) ? 5u : 6u;

        v4u g0;
        g0.x = 1u;                                     // count=1, no gather, user desc
        g0.y = lds_base;                               // lds_addr [63:32]
        g0.z = (unsigned)(ga & 0xffffffffull);         // global_addr [95:64]
        g0.w = (unsigned)((ga >> 32) & 0x01ffffffull)  // global_addr [120:96]
               | (2u << 30);                           // type=2 ("image")

        v8i g1;
        g1[0] = (int)((2u << 16)        // data_size = 4B
                      | (1u << 20)      // pad_enable
                      | (padcode << 22) // pad_interval = K dwords
                      | (3u << 25));    // pad_amount code 3 = 4 dwords
        g1[1] = (int)(((unsigned)K & 0xffffu) << 16);  // tensor_dim0 = K (low 16)
        g1[2] = (int)(16u << 16);                      // dim0 hi=0, tensor_dim1 = 16
        g1[3] = (int)(((unsigned)K & 0xffffu) << 16);  // dim1 hi=0, tile_dim0 = K
        g1[4] = 16;                                    // tile_dim1 = 16, tile_dim2 = 0
        g1[5] = K;                                     // tensor_dim0_stride = K
        g1[6] = 0;
        g1[7] = 0;

        v4i gz4 = {0, 0, 0, 0};                        // groups 2/3 unused (2D tensor)
        v8i gz8 = {0, 0, 0, 0, 0, 0, 0, 0};            // extra group (clang-23 6-arg form)
        __builtin_amdgcn_tensor_load_to_lds(g0, g1, gz4, gz4, gz8, 0);
        __builtin_amdgcn_s_wait_tensorcnt((short)0);
    }
    __syncthreads();   // publish the LDS tile to all 8 waves

    // --- WMMA main loop: A from LDS, B from global (L2-resident weights) ---
    v8f acc = {};
    const float* arow = &sA[(lane & 15) * RS + khalf];
    for (int k = 0; k < K; k += 4) {
        v2f a, b;
        // A-matrix 16x4 layout: VGPR0 = K=0|2, VGPR1 = K=1|3, rows across lanes
        a.x = arow[k];
        a.y = arow[k + 1];
        // B-matrix 4x16: column n = lane&15, K along VGPRs (mirror of A layout)
        b.x = B[(size_t)(k + khalf) * HID + col];
        b.y = B[(size_t)(k + khalf + 1) * HID + col];
        acc = __builtin_amdgcn_wmma_f32_16x16x4_f32(false, a, false, b, (short)0, acc,
                                                    false, false);
    }
    const float bv    = bias ? bias[col] : 0.f;
    const int   rbase = mbase + ((lane >> 4) << 3);   // M = r (lanes 0-15), r+8 (16-31)
#pragma unroll
    for (int r = 0; r < 8; ++r)
        C[(size_t)(rbase + r) * HID + col] = acc[r] + bv;
}

// ---------------------------------------------------------------------------
// e_src[n,h] = sum_d hW[n,h,d]*att_src[h,d] ; same for e_dst
// ---------------------------------------------------------------------------
__global__ void attn_coeff_kernel(const float* __restrict__ hW,
                                  const float* __restrict__ att_src,
                                  const float* __restrict__ att_dst,
                                  float* __restrict__ e_src, float* __restrict__ e_dst,
                                  int n_nodes) {
    int t = blockIdx.x * blockDim.x + threadIdx.x;
    if (t >= n_nodes * NHEAD) return;
    int n = t >> 3, h = t & 7;
    const float* hp = hW + (size_t)n * HID + h * DHEAD;
    const float* as = att_src + h * DHEAD;
    const float* ad = att_dst + h * DHEAD;
    float s = 0.f, d = 0.f;
#pragma unroll
    for (int i = 0; i < DHEAD; ++i) {
        float v = hp[i];
        s += v * as[i];
        d += v * ad[i];
    }
    e_src[t] = s;
    e_dst[t] = d;
}

// ---------------------------------------------------------------------------
// segment max of leaky_relu(e_src[s]+e_dst[d]) over incoming edges (atomic_max_u32)
// ---------------------------------------------------------------------------
__global__ void edge_max_kernel(const long long* __restrict__ src,
                                const long long* __restrict__ dst,
                                const float* __restrict__ e_src,
                                const float* __restrict__ e_dst,
                                unsigned* __restrict__ m_ord, int Etot, int Eraw) {
    int t = blockIdx.x * blockDim.x + threadIdx.x;
    if (t >= Etot * NHEAD) return;
    int e = t >> 3, h = t & 7;
    int s, d;
    if (e < Eraw) { s = (int)src[e]; d = (int)dst[e]; }
    else          { s = e - Eraw;    d = s; }         // self-loop
    float ev = lrelu(e_src[s * NHEAD + h] + e_dst[d * NHEAD + h]);
    atomicMax(&m_ord[d * NHEAD + h], ord_enc(ev));
}

// ---------------------------------------------------------------------------
// ex = exp(e - m[dst]); store per-edge, denom[dst] += ex
// ---------------------------------------------------------------------------
__global__ void edge_denom_kernel(const long long* __restrict__ src,
                                  const long long* __restrict__ dst,
                                  const float* __restrict__ e_src,
                                  const float* __restrict__ e_dst,
                                  const unsigned* __restrict__ m_ord,
                                  float* __restrict__ denom, float* __restrict__ exbuf,
                                  int Etot, int Eraw) {
    int t = blockIdx.x * blockDim.x + threadIdx.x;
    if (t >= Etot * NHEAD) return;
    int e = t >> 3, h = t & 7;
    int s, d;
    if (e < Eraw) { s = (int)src[e]; d = (int)dst[e]; }
    else          { s = e - Eraw;    d = s; }
    float ev = lrelu(e_src[s * NHEAD + h] + e_dst[d * NHEAD + h]);
    float m  = ord_dec(m_ord[d * NHEAD + h]);   // finite: every node has a self-loop
    float ex = __expf(ev - m);
    exbuf[t] = ex;
    atomicAdd(&denom[d * NHEAD + h], ex);
}

// ---------------------------------------------------------------------------
// outacc[d,:] += ex * hW[s,:]  — one wave per edge, 32-lane coalesced atomics
// ---------------------------------------------------------------------------
__global__ void edge_agg_kernel(const long long* __restrict__ src,
                                const long long* __restrict__ dst,
                                const float* __restrict__ hW,
                                const float* __restrict__ exbuf,
                                float* __restrict__ outacc, int Etot, int Eraw) {
    int lane = threadIdx.x & 31;
    int e    = blockIdx.x * (blockDim.x >> 5) + (threadIdx.x >> 5);
    if (e >= Etot) return;
    int s, d;
    if (e < Eraw) { s = (int)src[e]; d = (int)dst[e]; }
    else          { s = e - Eraw;    d = s; }
    const float* hs  = hW + (size_t)s * HID;
    float*       od  = outacc + (size_t)d * HID;
    const float* exb = exbuf + (size_t)e * NHEAD;
#pragma unroll
    for (int j = 0; j < 4; ++j) {
        int   c     = j * 32 + lane;
        float coeff = exb[c >> 4];
        atomicAdd(&od[c], coeff * hs[c]);
    }
}

// h = outacc/(denom+1e-16) + gat_b + h   (residual)
__global__ void finalize_attn_kernel(const float* __restrict__ outacc,
                                     const float* __restrict__ denom,
                                     const float* __restrict__ gat_b,
                                     float* __restrict__ h, int n_nodes) {
    int t = blockIdx.x * blockDim.x + threadIdx.x;
    if (t >= n_nodes * HID) return;
    int n = t >> 7, c = t & 127;
    float v = outacc[t] / (denom[n * NHEAD + (c >> 4)] + 1e-16f);
    h[t]    = v + gat_b[c] + h[t];
}

// ---------------------------------------------------------------------------
// GraphNorm: per-graph sums via LDS (ds_add_f32) then sparse global flush
// ---------------------------------------------------------------------------
__global__ void gn_stats1_kernel(const float* __restrict__ h,
                                 const long long* __restrict__ batch,
                                 float* __restrict__ gsum, float* __restrict__ gcnt,
                                 int n_nodes) {
    __shared__ float s_sum[NGRAPH * HID];
    __shared__ float s_cnt[NGRAPH];
    for (int j = threadIdx.x; j < NGRAPH * HID; j += blockDim.x) s_sum[j] = 0.f;
    if (threadIdx.x < NGRAPH) s_cnt[threadIdx.x] = 0.f;
    __syncthreads();
    int base = blockIdx.x * 128;  // 128 nodes per block
    for (int idx = threadIdx.x; idx < 128 * HID; idx += blockDim.x) {
        int nl = idx >> 7, c = idx & 127;
        int n  = base + nl;
        if (n < n_nodes) {
            int g = (int)batch[n];
            atomicAdd(&s_sum[g * HID + c], h[(size_t)n * HID + c]);
            if (c == 0) atomicAdd(&s_cnt[g], 1.f);
        }
    }
    __syncthreads();
    for (int j = threadIdx.x; j < NGRAPH * HID; j += blockDim.x)
        if (s_sum[j] != 0.f) atomicAdd(&gsum[j], s_sum[j]);
    if (threadIdx.x < NGRAPH && s_cnt[threadIdx.x] != 0.f)
        atomicAdd(&gcnt[threadIdx.x], s_cnt[threadIdx.x]);
}

__global__ void gn_mean_kernel(float* __restrict__ gsum, const float* __restrict__ gcnt) {
    int j = blockIdx.x * blockDim.x + threadIdx.x;
    if (j >= NGRAPH * HID) return;
    float c = gcnt[j >> 7];
    gsum[j] = gsum[j] / (c > 0.f ? c : 1.f);
}

// h -= scale*mean[g]; accumulate h^2 per graph
__global__ void gn_center_kernel(float* __restrict__ h, const long long* __restrict__ batch,
                                 const float* __restrict__ mean,
                                 const float* __restrict__ scale,
                                 float* __restrict__ gsumsq, int n_nodes) {
    __shared__ float s_sum[NGRAPH * HID];
    for (int j = threadIdx.x; j < NGRAPH * HID; j += blockDim.x) s_sum[j] = 0.f;
    __syncthreads();
    int base = blockIdx.x * 128;
    for (int idx = threadIdx.x; idx < 128 * HID; idx += blockDim.x) {
        int nl = idx >> 7, c = idx & 127;
        int n  = base + nl;
        if (n < n_nodes) {
            int   g = (int)batch[n];
            float v = h[(size_t)n * HID + c] - scale[c] * mean[g * HID + c];
            h[(size_t)n * HID + c] = v;
            atomicAdd(&s_sum[g * HID + c], v * v);
        }
    }
    __syncthreads();
    for (int j = threadIdx.x; j < NGRAPH * HID; j += blockDim.x)
        if (s_sum[j] != 0.f) atomicAdd(&gsumsq[j], s_sum[j]);
}

__global__ void gn_var_kernel(float* __restrict__ gsumsq, const float* __restrict__ gcnt) {
    int j = blockIdx.x * blockDim.x + threadIdx.x;
    if (j >= NGRAPH * HID) return;
    float c   = gcnt[j >> 7];
    float var = gsumsq[j] / (c > 0.f ? c : 1.f);
    gsumsq[j] = rsqrtf(var + EPSV);   // in-place: now holds 1/sqrt(var+eps)
}

__global__ void gn_apply_kernel(float* __restrict__ h, const long long* __restrict__ batch,
                                const float* __restrict__ inv, const float* __restrict__ w,
                                const float* __restrict__ b, int n_nodes) {
    int t = blockIdx.x * blockDim.x + threadIdx.x;
    if (t >= n_nodes * HID) return;
    int n = t >> 7, c = t & 127;
    int g = (int)batch[n];
    h[t]  = h[t] * inv[g * HID + c] * w[c] + b[c];
}

// ---------------------------------------------------------------------------
extern "C" void kernel_launch(void* const* d_in, const int* in_sizes, int n_in,
                              void* d_out, int out_size, void* d_ws, size_t ws_size,
                              hipStream_t stream) {
    (void)n_in; (void)out_size; (void)ws_size;
    const float*     x       = (const float*)d_in[0];
    const long long* eidx    = (const long long*)d_in[1];   // int64 per reference
    const long long* batch   = (const long long*)d_in[2];
    const float*     in_W    = (const float*)d_in[3];
    const float*     in_b    = (const float*)d_in[4];
    const float*     Wg      = (const float*)d_in[5];
    const float*     att_src = (const float*)d_in[6];
    const float*     att_dst = (const float*)d_in[7];
    const float*     gat_b   = (const float*)d_in[8];
    const float*     gn_w    = (const float*)d_in[9];
    const float*     gn_b    = (const float*)d_in[10];
    const float*     gn_s    = (const float*)d_in[11];

    const int n_nodes = in_sizes[0] / 64;       // 50000 (multiple of 16)
    const int Eraw    = in_sizes[1] / 2;        // 800000
    const int Etot    = Eraw + n_nodes;         // + self-loops
    const long long* src = eidx;
    const long long* dst = eidx + Eraw;

    // workspace carve-up
    char*  ws  = (char*)d_ws;
    size_t off = 0;
    auto alloc = [&](size_t bytes) -> void* {
        void* p = ws + off;
        off += (bytes + 255) & ~(size_t)255;
        return p;
    };
    float*    h      = (float*)alloc((size_t)n_nodes * HID * 4);
    float*    hW     = (float*)alloc((size_t)n_nodes * HID * 4);
    float*    outacc = (float*)alloc((size_t)n_nodes * HID * 4);
    float*    e_src  = (float*)alloc((size_t)n_nodes * NHEAD * 4);
    float*    e_dst  = (float*)alloc((size_t)n_nodes * NHEAD * 4);
    unsigned* m_ord  = (unsigned*)alloc((size_t)n_nodes * NHEAD * 4);
    float*    denom  = (float*)alloc((size_t)n_nodes * NHEAD * 4);
    float*    exbuf  = (float*)alloc((size_t)Etot * NHEAD * 4);
    float*    gsum   = (float*)alloc(NGRAPH * HID * 4);
    float*    gsumsq = (float*)alloc(NGRAPH * HID * 4);
    float*    gcnt   = (float*)alloc(NGRAPH * 4);

    const int nh   = n_nodes * NHEAD;
    const int eh   = Etot * NHEAD;
    const int nhid = n_nodes * HID;
    const int nblk = (n_nodes + 127) / 128;

    // input projection: h = x @ in_W + in_b  (K=64)
    gemm_wmma_f32<<<n_nodes / 16, 256, 0, stream>>>(x, in_W, in_b, h, 64);

    for (int l = 0; l < NLAYER; ++l) {
        // hW = h @ Wg[l]
        gemm_wmma_f32<<<n_nodes / 16, 256, 0, stream>>>(h, Wg + (size_t)l * HID * HID,
                                                        nullptr, hW, HID);
        attn_coeff_kernel<<<(nh + 255) / 256, 256, 0, stream>>>(
            hW, att_src + l * HID, att_dst + l * HID, e_src, e_dst, n_nodes);

        (void)hipMemsetAsync(m_ord, 0, (size_t)nh * 4, stream);
        (void)hipMemsetAsync(denom, 0, (size_t)nh * 4, stream);
        (void)hipMemsetAsync(outacc, 0, (size_t)nhid * 4, stream);

        edge_max_kernel<<<(eh + 255) / 256, 256, 0, stream>>>(src, dst, e_src, e_dst,
                                                              m_ord, Etot, Eraw);
        edge_denom_kernel<<<(eh + 255) / 256, 256, 0, stream>>>(
            src, dst, e_src, e_dst, m_ord, denom, exbuf, Etot, Eraw);
        edge_agg_kernel<<<(Etot + 7) / 8, 256, 0, stream>>>(src, dst, hW, exbuf, outacc,
                                                            Etot, Eraw);
        finalize_attn_kernel<<<(nhid + 255) / 256, 256, 0, stream>>>(
            outacc, denom, gat_b + l * HID, h, n_nodes);

        // GraphNorm
        (void)hipMemsetAsync(gsum, 0, NGRAPH * HID * 4, stream);
        (void)hipMemsetAsync(gcnt, 0, NGRAPH * 4, stream);
        gn_stats1_kernel<<<nblk, 256, 0, stream>>>(h, batch, gsum, gcnt, n_nodes);
        gn_mean_kernel<<<(NGRAPH * HID + 255) / 256, 256, 0, stream>>>(gsum, gcnt);
        (void)hipMemsetAsync(gsumsq, 0, NGRAPH * HID * 4, stream);
        gn_center_kernel<<<nblk, 256, 0, stream>>>(h, batch, gsum, gn_s + l * HID,
                                                   gsumsq, n_nodes);
        gn_var_kernel<<<(NGRAPH * HID + 255) / 256, 256, 0, stream>>>(gsumsq, gcnt);
        gn_apply_kernel<<<(nhid + 255) / 256, 256, 0, stream>>>(
            h, batch, gsumsq, gn_w + l * HID, gn_b + l * HID, n_nodes);
    }

    (void)hipMemcpyAsync(d_out, h, (size_t)nhid * 4, hipMemcpyDeviceToDevice, stream);
}